// GCNLayer_65137474011136
// MI455X (gfx1250) — compile-verified
//
#include <hip/hip_runtime.h>

typedef __attribute__((ext_vector_type(2))) float v2f;
typedef __attribute__((ext_vector_type(8))) float v8f;

#define NN 50000
#define NE 800000
#define IN_F 128
#define OUT_F 128

// ---------------------------------------------------------------------------
// 1) degree: deg[src[e]] += 1  (exact in f32 since deg < 2^24)
// ---------------------------------------------------------------------------
__global__ void deg_kernel(const long long* __restrict__ src, float* __restrict__ deg) {
    int e = blockIdx.x * blockDim.x + threadIdx.x;
    if (e < NE) atomicAdd(&deg[(int)src[e]], 1.0f);
}

// ---------------------------------------------------------------------------
// 2) dinv[n] = deg>0 ? rsqrt(deg) : 0   (in place)
// ---------------------------------------------------------------------------
__global__ void dinv_kernel(float* __restrict__ deg) {
    int n = blockIdx.x * blockDim.x + threadIdx.x;
    if (n < NN) {
        float d = deg[n];
        deg[n] = d > 0.0f ? rsqrtf(d) : 0.0f;
    }
}

// ---------------------------------------------------------------------------
// 3) agg[src] += x[dst] * dinv[src]*dinv[dst]
//    one wave32 per edge, 4 floats per lane (float4 gather + 4 f32 atomics)
// ---------------------------------------------------------------------------
__global__ void scatter_kernel(const long long* __restrict__ ei,
                               const float* __restrict__ x,
                               const float* __restrict__ dinv,
                               float* __restrict__ agg) {
    int tid  = blockIdx.x * blockDim.x + threadIdx.x;
    int e    = tid >> 5;
    int lane = tid & 31;
    if (e >= NE) return;
    int s = (int)ei[e];
    int d = (int)ei[NE + e];
    float norm = dinv[s] * dinv[d];
    const float4 v = ((const float4*)(x + (size_t)d * IN_F))[lane];
    float* a = agg + (size_t)s * IN_F + lane * 4;
    atomicAdd(a + 0, v.x * norm);
    atomicAdd(a + 1, v.y * norm);
    atomicAdd(a + 2, v.z * norm);
    atomicAdd(a + 3, v.w * norm);
}

// ---------------------------------------------------------------------------
// 4) out = relu(agg @ W^T + b) via V_WMMA_F32_16X16X4_F32
//    block = 128 thr = 4 waves; wave w -> N-tiles at n0 = 32w and 32w+16
//    M-tile = 16 rows per block (50000 = 3125 * 16, exact -> EXEC all-1s)
// ---------------------------------------------------------------------------
__global__ void __launch_bounds__(128)
gemm_kernel(const float* __restrict__ agg,   // [NN, 128]
            const float* __restrict__ W,     // [128, 128] (row = out feature)
            const float* __restrict__ bias,  // [128]
            float* __restrict__ out) {       // [NN, 128]
    const int wave = threadIdx.x >> 5;
    const int lane = threadIdx.x & 31;
    const int hi   = lane >> 4;       // 0: lanes 0-15, 1: lanes 16-31
    const int ml   = lane & 15;       // M (for A) / N (for B) within tile
    const int m0   = blockIdx.x * 16;
    const int n0   = wave * 32;       // this wave's first N-tile

    const float* arow = agg + (size_t)(m0 + ml) * IN_F;  // A: row m0+ml
    const float* w0   = W + (size_t)(n0 + ml) * IN_F;      // B tile 0: col n0+ml
    const float* w1   = W + (size_t)(n0 + 16 + ml) * IN_F; // B tile 1

    v8f c0 = {}, c1 = {};

    #pragma unroll
    for (int kk = 0; kk < IN_F / 4; ++kk) {
        const int kb = kk * 4 + 2 * hi;  // lanes 0-15: K+0/+1, lanes 16-31: K+2/+3
        v2f a, b0, b1;
        a.x  = arow[kb];  a.y  = arow[kb + 1];
        b0.x = w0[kb];    b0.y = w0[kb + 1];
        b1.x = w1[kb];    b1.y = w1[kb + 1];
        c0 = __builtin_amdgcn_wmma_f32_16x16x4_f32(false, a, false, b0,
                                                   (short)0, c0, false, false);
        c1 = __builtin_amdgcn_wmma_f32_16x16x4_f32(false, a, false, b1,
                                                   (short)0, c1, false, false);
    }

    // C/D layout: VGPR r -> M = r + 8*hi, N = ml
    const float bb0 = bias[n0 + ml];
    const float bb1 = bias[n0 + 16 + ml];
    float* orow = out + (size_t)(m0 + 8 * hi) * OUT_F;
    #pragma unroll
    for (int r = 0; r < 8; ++r) {
        orow[(size_t)r * OUT_F + n0 + ml]      = fmaxf(c0[r] + bb0, 0.0f);
        orow[(size_t)r * OUT_F + n0 + 16 + ml] = fmaxf(c1[r] + bb1, 0.0f);
    }
}

// ---------------------------------------------------------------------------
extern "C" void kernel_launch(void* const* d_in, const int* in_sizes, int n_in,
                              void* d_out, int out_size, void* d_ws, size_t ws_size,
                              hipStream_t stream) {
    const float*     x  = (const float*)d_in[0];
    const long long* ei = (const long long*)d_in[1];   // int64 [2, NE]
    const float*     W  = (const float*)d_in[2];
    const float*     b  = (const float*)d_in[3];
    float* out = (float*)d_out;

    float* deg = (float*)d_ws;          // NN floats
    float* agg = deg + NN;              // NN*128 floats (offset 200000 B, 16B aligned)

    hipMemsetAsync(d_ws, 0, (size_t)NN * (IN_F + 1) * sizeof(float), stream);

    deg_kernel<<<(NE + 255) / 256, 256, 0, stream>>>(ei, deg);
    dinv_kernel<<<(NN + 255) / 256, 256, 0, stream>>>(deg);
    scatter_kernel<<<((size_t)NE * 32 + 255) / 256, 256, 0, stream>>>(ei, x, deg, agg);
    gemm_kernel<<<NN / 16, 128, 0, stream>>>(agg, W, b, out);
}